// DynamicConvBlock_64965675319830
// MI455X (gfx1250) — compile-verified
//
#include <hip/hip_runtime.h>

typedef __attribute__((ext_vector_type(2))) float v2f;
typedef __attribute__((ext_vector_type(8))) float v8f;

#define Cc    192
#define COo   64
#define Hh    96
#define Ww    96
#define Bb    8
#define TW    16
#define THt   8
#define HALO  7
#define INW   (TW + 2*HALO)      // 30
#define INH   (THt + 2*HALO)     // 22
#define XS_ROW 31                // padded row stride (floats)
#define XS_CH  (INH * XS_ROW)    // 682
#define CBLK  16
#define NCB   (Cc / CBLK)        // 12
#define SS_STRIDE 17
#define NPX   (TW * THt)         // 128
#define WPT_STRIDE 65

// LDS float offsets
#define OFF_XS   0
#define OFF_SS   (OFF_XS  + CBLK * XS_CH)     // 10912
#define OFF_W7   (OFF_SS  + NPX * SS_STRIDE)  // 13088
#define OFF_WD5  (OFF_W7  + Cc * 49)          // 22496
#define OFF_WD7  (OFF_WD5 + Cc * 9)           // 24224
#define OFF_WPT  (OFF_WD7 + Cc * 9)           // 25952
#define LDS_FLOATS (OFF_WPT + Cc * WPT_STRIDE) // 38432 -> 153728 bytes

__global__ __launch_bounds__(256)
void dynconv_fused_wmma(const float* __restrict__ x,
                        const float* __restrict__ a,
                        const float* __restrict__ wk1,
                        const float* __restrict__ wk3,
                        const float* __restrict__ wk5,
                        const float* __restrict__ wk7,
                        const float* __restrict__ wk3d5,
                        const float* __restrict__ wk3d7,
                        const float* __restrict__ wp,
                        float* __restrict__ out)
{
    extern __shared__ float smem[];
    float* xs   = smem + OFF_XS;
    float* ss   = smem + OFF_SS;
    float* w7s  = smem + OFF_W7;
    float* wd5s = smem + OFF_WD5;
    float* wd7s = smem + OFF_WD7;
    float* wpT  = smem + OFF_WPT;

    const int t    = threadIdx.x;
    const int wave = t >> 5;
    const int lane = t & 31;

    // tile coordinates: 6 x-tiles, 12 y-tiles per image, 8 images
    const int blk = blockIdx.x;
    const int b   = blk / 72;
    const int rem = blk % 72;
    const int tyi = rem / 6;
    const int txi = rem % 6;
    const int gx0 = txi * TW;
    const int gy0 = tyi * THt;

    // ---------------- one-time weight preparation ----------------
    const float a0 = a[0], a1 = a[1], a2 = a[2], a3 = a[3];
    const float a4 = a[4], a5 = a[5], a6 = a[6], a7 = a[7];
    if (t < Cc) {
        const int ci = t;
        // combined dense 7x7 = a4*k7 + a3*k5 + (a2+a5)*k3 + center(a1*k1 + a0)
        for (int dy = 0; dy < 7; ++dy) {
            for (int dx = 0; dx < 7; ++dx) {
                float v = a4 * wk7[ci * 49 + dy * 7 + dx];
                if (dy >= 1 && dy <= 5 && dx >= 1 && dx <= 5)
                    v += a3 * wk5[ci * 25 + (dy - 1) * 5 + (dx - 1)];
                if (dy >= 2 && dy <= 4 && dx >= 2 && dx <= 4)
                    v += (a2 + a5) * wk3[ci * 9 + (dy - 2) * 3 + (dx - 2)];
                if (dy == 3 && dx == 3)
                    v += a1 * wk1[ci] + a0;
                w7s[ci * 49 + dy * 7 + dx] = v;
            }
        }
        for (int j = 0; j < 9; ++j) {
            wd5s[ci * 9 + j] = a6 * wk3d5[ci * 9 + j];
            wd7s[ci * 9 + j] = a7 * wk3d7[ci * 9 + j];
        }
        // transpose wp into [ci][co] for WMMA A-fragment gathers
        for (int co = 0; co < COo; ++co)
            wpT[ci * WPT_STRIDE + co] = wp[co * Cc + ci];
    }
    __syncthreads();

    // WMMA tile assignment: wave -> co tile (0..3) and 4 pixel-row tiles
    const int co_base = (wave & 3) * 16;
    const int r0      = (wave >> 2) * 4;

    v8f acc[4];
    #pragma unroll
    for (int i = 0; i < 4; ++i)
        #pragma unroll
        for (int j = 0; j < 8; ++j) acc[i][j] = 0.0f;

    const float* xbase = x + (size_t)b * Cc * Hh * Ww;

    for (int cb = 0; cb < NCB; ++cb) {
        const int ci0 = cb * CBLK;

        __syncthreads();   // previous WMMA phase done reading ss / xs reusable

        // ---- stage x tile (+halo, zero padded) into LDS ----
        for (int idx = t; idx < CBLK * INH * INW; idx += 256) {
            const int c  = idx / (INH * INW);
            const int r2 = idx - c * (INH * INW);
            const int y  = r2 / INW;
            const int xo = r2 - y * INW;
            const int gy = gy0 - HALO + y;
            const int gx = gx0 - HALO + xo;
            float v = 0.0f;
            if (gy >= 0 && gy < Hh && gx >= 0 && gx < Ww)
                v = xbase[(size_t)(ci0 + c) * (Hh * Ww) + gy * Ww + gx];
            xs[c * XS_CH + y * XS_ROW + xo] = v;
        }
        __syncthreads();

        // ---- depthwise (combined 7x7 + two dilated 3x3) + ReLU -> ss ----
        {
            const int ci   = t & 15;
            const int r    = (t >> 4) & 7;
            const int half = t >> 7;
            const int xoff = half * 8;
            const float* xc  = xs   + ci * XS_CH;
            const float* w7  = w7s  + ci * 49;
            const float* w5  = wd5s + ci * 9;
            const float* w7d = wd7s + ci * 9;

            float s8[8];
            #pragma unroll
            for (int i = 0; i < 8; ++i) s8[i] = 0.0f;

            // dense 7x7
            #pragma unroll
            for (int dy = 0; dy < 7; ++dy) {
                const float* row = xc + (r + 4 + dy) * XS_ROW + xoff + 4;
                float xv[14];
                #pragma unroll
                for (int j = 0; j < 14; ++j) xv[j] = row[j];
                #pragma unroll
                for (int dx = 0; dx < 7; ++dx) {
                    const float w = w7[dy * 7 + dx];
                    #pragma unroll
                    for (int i = 0; i < 8; ++i) s8[i] += w * xv[dx + i];
                }
            }
            // 3x3 dilation 5
            #pragma unroll
            for (int dy = 0; dy < 3; ++dy) {
                const float* row = xc + (r + 7 + 5 * (dy - 1)) * XS_ROW + xoff + 2;
                float xv[18];
                #pragma unroll
                for (int j = 0; j < 18; ++j) xv[j] = row[j];
                #pragma unroll
                for (int dx = 0; dx < 3; ++dx) {
                    const float w = w5[dy * 3 + dx];
                    #pragma unroll
                    for (int i = 0; i < 8; ++i) s8[i] += w * xv[5 * dx + i];
                }
            }
            // 3x3 dilation 7
            #pragma unroll
            for (int dy = 0; dy < 3; ++dy) {
                const float* row = xc + (r + 7 + 7 * (dy - 1)) * XS_ROW + xoff;
                float xv[22];
                #pragma unroll
                for (int j = 0; j < 22; ++j) xv[j] = row[j];
                #pragma unroll
                for (int dx = 0; dx < 3; ++dx) {
                    const float w = w7d[dy * 3 + dx];
                    #pragma unroll
                    for (int i = 0; i < 8; ++i) s8[i] += w * xv[7 * dx + i];
                }
            }
            // ReLU + stage [pixel][ci]
            #pragma unroll
            for (int i = 0; i < 8; ++i) {
                const int p = r * TW + xoff + i;
                ss[p * SS_STRIDE + ci] = s8[i] > 0.0f ? s8[i] : 0.0f;
            }
        }
        __syncthreads();

        // ---- pointwise GEMM chunk via V_WMMA_F32_16X16X4_F32 ----
        {
            const int m  = lane & 15;  // A: row M ; B: col N
            const int hi = lane >> 4;  // selects K={0,1} vs K={2,3}
            #pragma unroll
            for (int ks = 0; ks < 4; ++ks) {
                const int cia = ci0 + ks * 4 + hi * 2;
                v2f A;
                A.x = wpT[cia * WPT_STRIDE + co_base + m];
                A.y = wpT[(cia + 1) * WPT_STRIDE + co_base + m];
                const int kb = ks * 4 + hi * 2;
                #pragma unroll
                for (int rr = 0; rr < 4; ++rr) {
                    const int p = (r0 + rr) * TW + m;
                    v2f Bv;
                    Bv.x = ss[p * SS_STRIDE + kb];
                    Bv.y = ss[p * SS_STRIDE + kb + 1];
                    acc[rr] = __builtin_amdgcn_wmma_f32_16x16x4_f32(
                        false, A, false, Bv, (short)0, acc[rr], false, false);
                }
            }
        }
    }

    // ---------------- write D tiles ----------------
    {
        const int n  = lane & 15;
        const int hi = lane >> 4;
        float* obase = out + (size_t)b * COo * Hh * Ww;
        #pragma unroll
        for (int rr = 0; rr < 4; ++rr) {
            const int hg = gy0 + r0 + rr;
            #pragma unroll
            for (int v = 0; v < 8; ++v) {
                const int co = co_base + hi * 8 + v;
                obase[(size_t)co * (Hh * Ww) + hg * Ww + gx0 + n] = acc[rr][v];
            }
        }
    }
}

extern "C" void kernel_launch(void* const* d_in, const int* in_sizes, int n_in,
                              void* d_out, int out_size, void* d_ws, size_t ws_size,
                              hipStream_t stream) {
    (void)in_sizes; (void)n_in; (void)d_ws; (void)ws_size; (void)out_size;
    const float* x     = (const float*)d_in[0];
    const float* a     = (const float*)d_in[1];
    const float* wk1   = (const float*)d_in[2];
    const float* wk3   = (const float*)d_in[3];
    const float* wk5   = (const float*)d_in[4];
    const float* wk7   = (const float*)d_in[5];
    const float* wk3d5 = (const float*)d_in[6];
    const float* wk3d7 = (const float*)d_in[7];
    const float* wp    = (const float*)d_in[8];
    float* out = (float*)d_out;

    const dim3 grid(Bb * 12 * 6);   // 576 workgroups
    const dim3 block(256);
    const size_t shmem = (size_t)LDS_FLOATS * sizeof(float); // 153,728 B

    hipLaunchKernelGGL(dynconv_fused_wmma, grid, block, shmem, stream,
                       x, a, wk1, wk3, wk5, wk7, wk3d5, wk3d7, wp, out);
}